// FVModel_1185410974250
// MI455X (gfx1250) — compile-verified
//
#include <hip/hip_runtime.h>
#include <hip/hip_bf16.h>

typedef __attribute__((ext_vector_type(2))) float v2f;
typedef __attribute__((ext_vector_type(8))) float v8f;

#define BB 128
#define TT 50
#define DD 144       // E + EW
#define EE 128
#define GG 432       // 3*D

// ---------------------------------------------------------------------------
// Kernel 1: basket embedding (mean-pool of gathered rows + flag embedding)
// grid = B*T blocks, 128 threads (one per embedding dim)
// ---------------------------------------------------------------------------
__global__ void embed_kernel(const int* __restrict__ x,
                             const float* __restrict__ encode_w,
                             const float* __restrict__ wchange_w,
                             float* __restrict__ seq) {
  const int bt = blockIdx.x;            // b*T + t
  const int e  = threadIdx.x;           // 0..127
  const int* row = x + bt * 33;         // 32 items + flag
  float acc = 0.f;
  int cnt = 0;
  for (int j = 0; j < 32; ++j) {
    const int it = row[j];
    if (it != 0) {
      acc += encode_w[(size_t)it * EE + e];
      ++cnt;
    }
  }
  const float pooled = acc / (float)(cnt < 1 ? 1 : cnt);
  seq[(size_t)bt * DD + e] = pooled;
  if (e < 16) {
    const int flag = row[32];
    seq[(size_t)bt * DD + EE + e] = wchange_w[flag * 16 + e];
  }
}

// ---------------------------------------------------------------------------
// Generic fp32 WMMA GEMM:  C[M,N] = A[M,K] @ W^T + bias,   W is (N,K) row-major
// M,N multiples of 16; K multiple of 4. One wave per 16x16 output tile.
// ---------------------------------------------------------------------------
__global__ void wmma_gemm_bias(const float* __restrict__ A,
                               const float* __restrict__ W,
                               const float* __restrict__ bias,
                               float* __restrict__ C,
                               int M, int N, int K) {
  const int wave = threadIdx.x >> 5;
  const int lane = threadIdx.x & 31;
  const int tilesPerRow = N >> 4;
  const int tile = blockIdx.x * (blockDim.x >> 5) + wave;
  const int ntiles = (M >> 4) * tilesPerRow;
  if (tile >= ntiles) return;                    // whole-wave uniform exit
  const int m0 = (tile / tilesPerRow) << 4;
  const int n0 = (tile % tilesPerRow) << 4;
  const int half = lane >> 4;
  const int lr   = lane & 15;

  v8f c = {};
  const float* __restrict__ arow = A + (size_t)(m0 + lr) * K;
  const float* __restrict__ wrow = W + (size_t)(n0 + lr) * K;
  for (int kb = 0; kb < K; kb += 4) {
    const int k = kb + 2 * half;
    v2f a; a.x = arow[k]; a.y = arow[k + 1];     // A[row=lr][k..k+1]
    v2f b; b.x = wrow[k]; b.y = wrow[k + 1];     // B^T: W[col=lr][k..k+1]
    c = __builtin_amdgcn_wmma_f32_16x16x4_f32(false, a, false, b,
                                              (short)0, c, false, false);
  }
  const float bn = bias ? bias[n0 + lr] : 0.f;
#pragma unroll
  for (int v = 0; v < 8; ++v)
    C[(size_t)(m0 + v + 8 * half) * N + n0 + lr] = c[v] + bn;
}

// ---------------------------------------------------------------------------
// Kernel 3: fused recurrent GRU step.
// Block = 96 threads = 3 waves; wave g computes the 16x16 GH tile of gate g
// (r/z/n) via WMMA over K=144, exchanges through LDS, then all threads apply
// the gate nonlinearity for a 16(row) x 16(d) patch.
// grid = (B/16) * (D/16) = 8*9 = 72 blocks.
// ---------------------------------------------------------------------------
__global__ void gru_step(const float* __restrict__ GI,     // (B*T, 432)
                         const float* __restrict__ w_hh,   // (432, 144)
                         const float* __restrict__ b_hh,   // (432)
                         const float* __restrict__ h_in,   // (B, 144)
                         float* __restrict__ h_out,        // (B, 144)
                         float* __restrict__ outs,         // (B, T, 144)
                         int t) {
  __shared__ float gh[3][256];
  const int wave  = threadIdx.x >> 5;           // gate index 0..2
  const int lane  = threadIdx.x & 31;
  const int mtile = blockIdx.x / 9;
  const int dtile = blockIdx.x % 9;
  const int m0 = mtile << 4;
  const int d0 = dtile << 4;
  const int n0 = wave * DD + d0;                // column block inside GH
  const int half = lane >> 4;
  const int lr   = lane & 15;

  v8f c = {};
  const float* __restrict__ hrow = h_in + (size_t)(m0 + lr) * DD;
  const float* __restrict__ wrow = w_hh + (size_t)(n0 + lr) * DD;
#pragma unroll 4
  for (int kb = 0; kb < DD; kb += 4) {
    const int k = kb + 2 * half;
    v2f a; a.x = hrow[k]; a.y = hrow[k + 1];
    v2f b; b.x = wrow[k]; b.y = wrow[k + 1];
    c = __builtin_amdgcn_wmma_f32_16x16x4_f32(false, a, false, b,
                                              (short)0, c, false, false);
  }
  const float bn = b_hh[n0 + lr];
#pragma unroll
  for (int v = 0; v < 8; ++v)
    gh[wave][(v + 8 * half) * 16 + lr] = c[v] + bn;
  __syncthreads();

  for (int i = threadIdx.x; i < 256; i += 96) {
    const int row = i >> 4;
    const int col = i & 15;
    const int b = m0 + row;
    const int d = d0 + col;
    const float* gi = GI + ((size_t)b * TT + t) * GG + d;
    const float gir = gi[0], giz = gi[DD], gin = gi[2 * DD];
    const float ghr = gh[0][i], ghz = gh[1][i], ghn = gh[2][i];
    const float r = 1.f / (1.f + __expf(-(gir + ghr)));
    const float z = 1.f / (1.f + __expf(-(giz + ghz)));
    const float n = tanhf(gin + r * ghn);
    const float hprev = h_in[(size_t)b * DD + d];
    const float hn = (1.f - z) * n + z * hprev;
    h_out[(size_t)b * DD + d] = hn;
    outs[((size_t)b * TT + t) * DD + d] = hn;
  }
}

// ---------------------------------------------------------------------------
extern "C" void kernel_launch(void* const* d_in, const int* in_sizes, int n_in,
                              void* d_out, int out_size, void* d_ws, size_t ws_size,
                              hipStream_t stream) {
  const int*   x         = (const int*)  d_in[0];
  // d_in[1] = lengths (int64, unused: all == T)
  const float* hidden    = (const float*)d_in[2];   // (1,B,D) zeros-init state
  const float* encode_w  = (const float*)d_in[3];
  const float* wchange_w = (const float*)d_in[4];
  const float* w_ih      = (const float*)d_in[5];
  const float* w_hh      = (const float*)d_in[6];
  const float* b_ih      = (const float*)d_in[7];
  const float* b_hh      = (const float*)d_in[8];
  const float* fc_w      = (const float*)d_in[9];
  const float* fc_b      = (const float*)d_in[10];

  // workspace layout (floats)
  float* ws   = (float*)d_ws;
  float* seq  = ws;                          // 6400*144
  float* GI   = seq  + (size_t)BB * TT * DD; // 6400*432
  float* outs = GI   + (size_t)BB * TT * GG; // 6400*144
  float* h0   = outs + (size_t)BB * TT * DD; // 128*144
  float* h1   = h0   + (size_t)BB * DD;      // 128*144

  float* dyn       = (float*)d_out;                       // (B,T,E)
  float* hlast_out = dyn + (size_t)BB * TT * EE;          // (1,B,D)

  // 1) embed baskets -> seq (B*T, 144)
  embed_kernel<<<BB * TT, EE, 0, stream>>>(x, encode_w, wchange_w, seq);

  // 2) hoisted input projection: GI = seq @ w_ih^T + b_ih   (6400x432x144)
  //    tiles = 400 * 27 = 10800; 4 waves / block -> 2700 blocks
  wmma_gemm_bias<<<2700, 128, 0, stream>>>(seq, w_ih, b_ih, GI,
                                           BB * TT, GG, DD);

  // 3) init h ping-pong buffer from input hidden state
  hipMemcpyAsync(h0, hidden, (size_t)BB * DD * sizeof(float),
                 hipMemcpyDeviceToDevice, stream);

  // 4) serial recurrence: one fused WMMA+gates kernel per timestep
  float* hcur = h0;
  float* hnxt = h1;
  for (int t = 0; t < TT; ++t) {
    gru_step<<<72, 96, 0, stream>>>(GI, w_hh, b_hh, hcur, hnxt, outs, t);
    float* tmp = hcur; hcur = hnxt; hnxt = tmp;
  }

  // 5) FC epilogue: dyn = outs @ fc_w^T + fc_b   (6400x128x144)
  //    tiles = 400 * 8 = 3200; 4 waves / block -> 800 blocks
  wmma_gemm_bias<<<800, 128, 0, stream>>>(outs, fc_w, fc_b, dyn,
                                          BB * TT, EE, DD);

  // 6) h_last (after even number of swaps hcur holds final state)
  hipMemcpyAsync(hlast_out, hcur, (size_t)BB * DD * sizeof(float),
                 hipMemcpyDeviceToDevice, stream);
}